// PhysicsInformedLoss_84963043050069
// MI455X (gfx1250) — compile-verified
//
#include <hip/hip_runtime.h>
#include <math.h>

// ---- problem constants (from reference) ----
#define Wd   128
#define Hd   128
#define Dd   64
#define Bn   8
#define N_TOT (Bn * Dd * Hd * Wd)     // 8388608 elements
#define NV4   (N_TOT / 4)             // 2097152 float4 quads
#define NBLOCKS 2048
#define BLOCK   256                    // 8 waves (wave32)

typedef __attribute__((ext_vector_type(2))) float v2f;
typedef __attribute__((ext_vector_type(4))) float v4f;
typedef __attribute__((ext_vector_type(8))) float v8f;

// Wave32 full-precision sum via V_WMMA_F32_16X16X4_F32.
// A[m][0] = acc(lane m), A[m][2] = acc(lane m+16), A[m][1]=A[m][3]=0.
// B = all ones  =>  D[m][n] = acc[m] + acc[m+16]  (independent of n).
// Lane n<16 holds rows 0..7 in its 8 D regs; lane n>=16 holds rows 8..15.
// Summing the 8 regs then xor-16 shuffle yields the full 32-lane sum.
// Requires EXEC all ones (call sites have uniform control flow).
__device__ __forceinline__ float wave_sum_wmma(float acc) {
    v2f a; a[0] = acc;  a[1] = 0.0f;
    v2f b; b[0] = 1.0f; b[1] = 1.0f;
    v8f c = {};
    c = __builtin_amdgcn_wmma_f32_16x16x4_f32(
            /*neg_a=*/false, a, /*neg_b=*/false, b,
            /*c_mod=*/(short)0, c, /*reuse_a=*/false, /*reuse_b=*/false);
    float s = ((c[0] + c[1]) + (c[2] + c[3])) + ((c[4] + c[5]) + (c[6] + c[7]));
    s += __shfl_xor(s, 16, 32);
    return s;
}

__global__ __launch_bounds__(BLOCK)
void pil_main_kernel(const float* __restrict__ Tp,   // T_pred
                     const float* __restrict__ Tt,   // T_target
                     const float* __restrict__ Tin,  // T_in
                     const float* __restrict__ Q,
                     const float* __restrict__ dtp,  // scalar dt
                     const float* __restrict__ Kf,   // k_field
                     const float* __restrict__ Cp,
                     const float* __restrict__ Rho,
                     const float* __restrict__ Mp,   // mask_pred (logits)
                     const float* __restrict__ Mt,   // mask_target
                     float* __restrict__ partial)    // [NBLOCKS*3]
{
    const v4f* __restrict__ Tp4  = (const v4f*)Tp;
    const v4f* __restrict__ Kf4  = (const v4f*)Kf;
    const v4f* __restrict__ Tt4  = (const v4f*)Tt;
    const v4f* __restrict__ Tin4 = (const v4f*)Tin;
    const v4f* __restrict__ Q4   = (const v4f*)Q;
    const v4f* __restrict__ Cp4  = (const v4f*)Cp;
    const v4f* __restrict__ Rho4 = (const v4f*)Rho;
    const v4f* __restrict__ Mp4  = (const v4f*)Mp;
    const v4f* __restrict__ Mt4  = (const v4f*)Mt;

    const float inv_dt = 1.0f / dtp[0];
    const float inv_d  = 1.0f / 5e-5f;      // 1/dx = 1/dy = 1/dz
    const float inv_qr = 1e-12f;            // 1/Q_REF

    float s_mse = 0.0f, s_pde = 0.0f, s_bce = 0.0f;

    const int stride = NBLOCKS * BLOCK;     // NV4 / stride == 4 exact iterations
    for (int v = blockIdx.x * BLOCK + threadIdx.x; v < NV4; v += stride) {
        const int base = v << 2;
        const int wb = base & (Wd - 1);          // 0,4,...,124 (same for the quad)
        const int h  = (v >> 5)  & (Hd - 1);     // uniform over the quad
        const int d  = (v >> 12) & (Dd - 1);     // uniform over the quad

        // center quad
        const v4f t  = Tp4[v];
        const v4f kk = Kf4[v];

        // x-edge scalars, clamped-to-self at domain boundary (=> exact zero flux)
        const int im = base + ((wb == 0)        ? 0 : -1);
        const int ip = base + ((wb == Wd - 4)   ? 3 :  4);
        const float t_m1 = Tp[im], k_m1 = Kf[im];
        const float t_p4 = Tp[ip], k_p4 = Kf[ip];

        // y/z neighbor quads, clamped-to-self at domain boundary
        const int vyl = v + ((h == 0)      ? 0 : -(Wd / 4));
        const int vyh = v + ((h == Hd - 1) ? 0 :  (Wd / 4));
        const int vzl = v + ((d == 0)      ? 0 : -(Wd * Hd / 4));
        const int vzh = v + ((d == Dd - 1) ? 0 :  (Wd * Hd / 4));
        const v4f tyl = Tp4[vyl], kyl = Kf4[vyl];
        const v4f tyh = Tp4[vyh], kyh = Kf4[vyh];
        const v4f tzl = Tp4[vzl], kzl = Kf4[vzl];
        const v4f tzh = Tp4[vzh], kzh = Kf4[vzh];

        // single-use fields: non-temporal b128 loads (preserve stencil lines in cache)
        const v4f tt  = __builtin_nontemporal_load(&Tt4[v]);
        const v4f tin = __builtin_nontemporal_load(&Tin4[v]);
        const v4f q   = __builtin_nontemporal_load(&Q4[v]);
        const v4f cp  = __builtin_nontemporal_load(&Cp4[v]);
        const v4f rho = __builtin_nontemporal_load(&Rho4[v]);
        const v4f xm  = __builtin_nontemporal_load(&Mp4[v]);
        const v4f ym  = __builtin_nontemporal_load(&Mt4[v]);

        #pragma unroll
        for (int j = 0; j < 4; ++j) {
            const float tc = t[j];
            const float kc = kk[j];
            const float txm = (j == 0) ? t_m1 : t[j - 1];
            const float kxm = (j == 0) ? k_m1 : kk[j - 1];
            const float txp = (j == 3) ? t_p4 : t[j + 1];
            const float kxp = (j == 3) ? k_p4 : kk[j + 1];

            // conservative flux form, matching reference: (/d) on flux, (/d) on div
            float fxh = 0.5f * (kc + kxp)   * (txp    - tc) * inv_d;
            float fxl = 0.5f * (kxm + kc)   * (tc  - txm)   * inv_d;
            float div = (fxh - fxl) * inv_d;
            float fyh = 0.5f * (kc + kyh[j]) * (tyh[j] - tc) * inv_d;
            float fyl = 0.5f * (kyl[j] + kc) * (tc - tyl[j]) * inv_d;
            div += (fyh - fyl) * inv_d;
            float fzh = 0.5f * (kc + kzh[j]) * (tzh[j] - tc) * inv_d;
            float fzl = 0.5f * (kzl[j] + kc) * (tc - tzl[j]) * inv_d;
            div += (fzh - fzl) * inv_d;

            const float diff = tc - tt[j];
            s_mse += diff * diff;

            const float dTdt  = (tc - tin[j]) * inv_dt;
            const float resid = rho[j] * cp[j] * dTdt - div - q[j];
            const float r     = resid * inv_qr;
            s_pde += r * r;

            // stable BCE-with-logits: max(x,0) - x*y + log1p(exp(-|x|))
            const float x = xm[j], y = ym[j];
            s_bce += fmaxf(x, 0.0f) - x * y + log1pf(expf(-fabsf(x)));
        }
    }

    // wave-level exact fp32 sums via WMMA (all 256 threads active here)
    const float w0 = wave_sum_wmma(s_mse);
    const float w1 = wave_sum_wmma(s_pde);
    const float w2 = wave_sum_wmma(s_bce);

    __shared__ float lds[BLOCK / 32][3];
    const int lane = threadIdx.x & 31;
    const int wv   = threadIdx.x >> 5;
    if (lane == 0) { lds[wv][0] = w0; lds[wv][1] = w1; lds[wv][2] = w2; }
    __syncthreads();
    if (threadIdx.x == 0) {
        float a = 0.0f, b = 0.0f, c = 0.0f;
        #pragma unroll
        for (int i = 0; i < BLOCK / 32; ++i) { a += lds[i][0]; b += lds[i][1]; c += lds[i][2]; }
        partial[blockIdx.x * 3 + 0] = a;
        partial[blockIdx.x * 3 + 1] = b;
        partial[blockIdx.x * 3 + 2] = c;
    }
}

__global__ __launch_bounds__(BLOCK)
void pil_final_kernel(const float* __restrict__ partial, float* __restrict__ out)
{
    float a = 0.0f, b = 0.0f, c = 0.0f;
    for (int i = threadIdx.x; i < NBLOCKS; i += BLOCK) {
        a += partial[i * 3 + 0];
        b += partial[i * 3 + 1];
        c += partial[i * 3 + 2];
    }
    const float w0 = wave_sum_wmma(a);
    const float w1 = wave_sum_wmma(b);
    const float w2 = wave_sum_wmma(c);

    __shared__ float lds[BLOCK / 32][3];
    const int lane = threadIdx.x & 31;
    const int wv   = threadIdx.x >> 5;
    if (lane == 0) { lds[wv][0] = w0; lds[wv][1] = w1; lds[wv][2] = w2; }
    __syncthreads();
    if (threadIdx.x == 0) {
        float sa = 0.0f, sb = 0.0f, sc = 0.0f;
        #pragma unroll
        for (int i = 0; i < BLOCK / 32; ++i) { sa += lds[i][0]; sb += lds[i][1]; sc += lds[i][2]; }
        const float invN = 1.0f / (float)N_TOT;
        const float mse = sa * invN * (1.0f / (2000.0f * 2000.0f)); // / T_REF^2
        const float pde = sb * invN;                                 // already /Q_REF^2
        const float bce = sc * invN;
        out[0] = mse + 0.1f * pde + 1.0f * bce;  // total (PDE_WEIGHT=0.1, MASK_WEIGHT=1)
        out[1] = mse;
        out[2] = pde;
        out[3] = bce;
    }
}

extern "C" void kernel_launch(void* const* d_in, const int* in_sizes, int n_in,
                              void* d_out, int out_size, void* d_ws, size_t ws_size,
                              hipStream_t stream) {
    (void)in_sizes; (void)n_in; (void)out_size; (void)ws_size;
    const float* Tp  = (const float*)d_in[0];
    const float* Tt  = (const float*)d_in[1];
    const float* Tin = (const float*)d_in[2];
    const float* Q   = (const float*)d_in[3];
    const float* dt  = (const float*)d_in[4];
    const float* Kf  = (const float*)d_in[5];
    const float* Cp  = (const float*)d_in[6];
    const float* Rho = (const float*)d_in[7];
    const float* Mp  = (const float*)d_in[8];
    const float* Mt  = (const float*)d_in[9];

    float* partial = (float*)d_ws;               // needs NBLOCKS*3*4 = 24 KB

    pil_main_kernel<<<NBLOCKS, BLOCK, 0, stream>>>(Tp, Tt, Tin, Q, dt, Kf, Cp, Rho,
                                                   Mp, Mt, partial);
    pil_final_kernel<<<1, BLOCK, 0, stream>>>(partial, (float*)d_out);
}